// HeteroGNN_59923383714577
// MI455X (gfx1250) — compile-verified
//
#include <hip/hip_runtime.h>

// ---------------------------------------------------------------------------
// HeteroGNN on MI455X (gfx1250). Dense blocked GEMM formulation of the regular
// graph (32 packets/flow, 1 flow/graph). fp32 WMMA (V_WMMA_F32_16X16X4_F32)
// preserves reference numerics (workload is memory-bound: ~15 GFLOP vs
// ~200 MB of mostly-L2-resident traffic at 23.3 TB/s). CDNA5 data movement:
// GLOBAL_LOAD_ASYNC_TO_LDS_B128 (ASYNCcnt) for activation/weight staging and
// TENSOR_LOAD_TO_LDS (TDM, TENSORcnt, 6-arg clang-23 form) for weight tiles.
// ---------------------------------------------------------------------------

typedef __attribute__((ext_vector_type(2))) float v2f;
typedef __attribute__((ext_vector_type(8))) float v8f;
typedef __attribute__((ext_vector_type(4))) unsigned int u32x4;
typedef __attribute__((ext_vector_type(8))) int i32x8;
typedef __attribute__((ext_vector_type(4))) int i32x4;

#define N_GRAPHS 16384
#define N_PKT    (N_GRAPHS * 32)
#define F_FLOW   76
#define F_PKT    32
#define HD       64
#define XPAD     36   // LDS row pitch for x_packet tiles (16B aligned, bank-spread)

static __device__ __forceinline__ v8f wmma4(float a0, float a1, float b0, float b1, v8f c) {
  v2f a; a.x = a0; a.y = a1;
  v2f b; b.x = b0; b.y = b1;
  return __builtin_amdgcn_wmma_f32_16x16x4_f32(false, a, false, b, (short)0, c, false, false);
}

static __device__ __forceinline__ float lrelu(float x) { return x > 0.f ? x : 0.01f * x; }

// ---- CDNA5 async global->LDS copy (per-lane 16B), tracked by ASYNCcnt ----
static __device__ __forceinline__ void async_copy_b128(const float* gsrc, float* ldst) {
  asm volatile("global_load_async_to_lds_b128 %0, %1, off"
               :: "v"((__attribute__((address_space(3))) float*)ldst),
                  "v"((const __attribute__((address_space(1))) float*)gsrc)
               : "memory");
}
static __device__ __forceinline__ void async_wait0() {
#if __has_builtin(__builtin_amdgcn_s_wait_asynccnt)
  __builtin_amdgcn_s_wait_asynccnt(0);
#else
  asm volatile("s_wait_asynccnt 0" ::: "memory");
#endif
}

// ---- CDNA5 Tensor Data Mover: 2D tile (rows x cols fp32) global -> LDS ----
#if __has_builtin(__builtin_amdgcn_tensor_load_to_lds)
static __device__ __forceinline__ void tdm_load_2d(const float* gsrc, float* lds,
                                                   int rows, int cols) {
  unsigned long long ga = (unsigned long long)(uintptr_t)gsrc;
  unsigned lo = (unsigned)(uintptr_t)(__attribute__((address_space(3))) float*)lds;
  u32x4 g0;
  g0[0] = 1u;                                   // count=1, user mode
  g0[1] = lo;                                   // lds_addr (bytes)
  g0[2] = (unsigned)ga;                         // global_addr[31:0]
  g0[3] = (unsigned)(ga >> 32) | 0x80000000u;   // global_addr[56:32] | type=2
  i32x8 g1;
  g1[0] = 0x20000;                              // data_size=4B, no multicast
  g1[1] = (cols & 0xffff) << 16;                // tensor_dim0[15:0]
  g1[2] = ((cols >> 16) & 0xffff) | ((rows & 0xffff) << 16);  // td0 hi | td1 lo
  g1[3] = ((rows >> 16) & 0xffff) | ((cols & 0xffff) << 16);  // td1 hi | tile_dim0
  g1[4] = rows & 0xffff;                        // tile_dim1 (tile_dim2=0)
  g1[5] = cols;                                 // tensor_dim0_stride[31:0]
  g1[6] = 0; g1[7] = 0;
  i32x4 z; z[0] = 0; z[1] = 0; z[2] = 0; z[3] = 0;
  i32x8 z8; z8[0] = 0; z8[1] = 0; z8[2] = 0; z8[3] = 0;
  z8[4] = 0; z8[5] = 0; z8[6] = 0; z8[7] = 0;
  // clang-23 / therock form: (g0, g1, g2, g3, extra, cpol)
  __builtin_amdgcn_tensor_load_to_lds(g0, g1, z, z, z8, 0);
}
#endif
static __device__ __forceinline__ void tensor_wait0() {
#if __has_builtin(__builtin_amdgcn_s_wait_tensorcnt)
  __builtin_amdgcn_s_wait_tensorcnt(0);
#else
  asm volatile("s_wait_tensorcnt 0" ::: "memory");
#endif
}

// --------------------------- utility kernels -------------------------------

__global__ __launch_bounds__(256) void zero_f32(float* p, int n) {
  int i = blockIdx.x * 256 + threadIdx.x;
  if (i < n) p[i] = 0.f;
}

// pm[g][c] = mean over 32 packet rows of x_packet (16384 x 32)
__global__ __launch_bounds__(256) void pkt_mean32(const float* __restrict__ xp,
                                                  float* __restrict__ pm) {
  int i = blockIdx.x * 256 + threadIdx.x;
  int g = i >> 5, c = i & 31;
  const float* p = xp + (long)g * 32 * F_PKT + c;
  float s = 0.f;
  #pragma unroll
  for (int r = 0; r < 32; ++r) s += p[r * F_PKT];
  pm[i] = s * (1.0f / 32.0f);
}

// ------------------------ generic WMMA GEMM (N = 64) -----------------------
// C[M x 64] (+)= A[M x K] @ B[K x 64] (+ bias). 8 waves/block, wave = 16x16 tile.
__global__ __launch_bounds__(256) void gemm_n64(const float* __restrict__ A, int lda,
                                                const float* __restrict__ B,
                                                const float* __restrict__ bias,
                                                float* __restrict__ C,
                                                int K, int accumulate) {
  __shared__ alignas(16) float sB[128 * 16];
  int tid = threadIdx.x, lane = tid & 31, wv = tid >> 5;
  int n0 = blockIdx.y * 16;
  int m0 = (blockIdx.x * 8 + wv) * 16;
  // async-stage the B column slice (K x 16) into LDS
  for (int i = tid; i < K * 4; i += 256) {
    int r = i >> 2, p = i & 3;
    async_copy_b128(B + r * 64 + n0 + p * 4, &sB[i * 4]);
  }
  async_wait0();
  __syncthreads();
  int arow = lane & 15, khalf = (lane >> 4) << 1, col = lane & 15;
  const float* Ar = A + (long)(m0 + arow) * lda;
  v8f acc = {};
  for (int k0 = 0; k0 < K; k0 += 4) {
    int kk = k0 + khalf;
    acc = wmma4(Ar[kk], Ar[kk + 1], sB[kk * 16 + col], sB[(kk + 1) * 16 + col], acc);
  }
  int rbase = m0 + ((lane >> 4) << 3);
  float bv = bias ? bias[n0 + col] : 0.f;
  #pragma unroll
  for (int v = 0; v < 8; ++v) {
    long idx = (long)(rbase + v) * 64 + n0 + col;
    float out = acc[v] + bv;
    if (accumulate) out += C[idx];
    C[idx] = out;
  }
}

// --------------------- packet pass 1: BN1 stats of k_raw -------------------
// k_raw = x_packet @ c1_fp_Wr + fl1[g]; per-block channel partials -> PP1.
__global__ __launch_bounds__(256) void pkt_pass1(const float* __restrict__ xp,
                                                 const float* __restrict__ W,   // 32x64
                                                 const float* __restrict__ fl1, // 16384x64
                                                 float* __restrict__ PP1) {
  __shared__ alignas(16) float sB[32 * 16];
  __shared__ alignas(16) float sX[128 * XPAD];
  __shared__ float sSl[8 * 32], sQl[8 * 32];
  int tid = threadIdx.x, lane = tid & 31, wv = tid >> 5;
  int n0 = blockIdx.y * 16;
  int m0 = (blockIdx.x * 8 + wv) * 16;
  // async-stage weight slice + this block's 128-row x_packet slab
  for (int i = tid; i < 32 * 4; i += 256) {
    int r = i >> 2, p = i & 3;
    async_copy_b128(W + r * 64 + n0 + p * 4, &sB[i * 4]);
  }
  {
    const float* xpb = xp + (long)blockIdx.x * 128 * F_PKT;
    for (int i = tid; i < 128 * 8; i += 256) {
      int r = i >> 3, c = i & 7;
      async_copy_b128(xpb + r * F_PKT + c * 4, &sX[r * XPAD + c * 4]);
    }
  }
  async_wait0();
  __syncthreads();
  int arow = lane & 15, khalf = (lane >> 4) << 1, col = lane & 15;
  const float* ArL = &sX[(wv * 16 + arow) * XPAD];
  v8f acc = {};
  #pragma unroll
  for (int k0 = 0; k0 < 32; k0 += 4) {
    int kk = k0 + khalf;
    acc = wmma4(ArL[kk], ArL[kk + 1], sB[kk * 16 + col], sB[(kk + 1) * 16 + col], acc);
  }
  int g = m0 >> 5;
  float fb = fl1[g * 64 + n0 + col];
  float s = 0.f, q = 0.f;
  #pragma unroll
  for (int v = 0; v < 8; ++v) { float x = acc[v] + fb; s += x; q += x * x; }
  sSl[wv * 32 + lane] = s; sQl[wv * 32 + lane] = q;
  __syncthreads();
  if (tid < 16) {
    float ts = 0.f, tq = 0.f;
    for (int w = 0; w < 8; ++w) {
      ts += sSl[w * 32 + tid] + sSl[w * 32 + tid + 16];
      tq += sQl[w * 32 + tid] + sQl[w * 32 + tid + 16];
    }
    int slot = blockIdx.y * 4096 + blockIdx.x;
    PP1[slot * 64 + n0 + tid] = ts;
    PP1[16384 * 64 + slot * 64 + n0 + tid] = tq;
  }
}

// ------------- packet passes 2/3: layer-2 GEMM chain (recompute) -----------
__global__ __launch_bounds__(128) void pkt_pass23(
    const float* __restrict__ xp,
    const float* __restrict__ W1,      // 32x64  c1_fp_Wr
    const float* __restrict__ W2,      // 64x64  c2_fp_Wr
    const float* __restrict__ fl1,
    const float* __restrict__ psc1, const float* __restrict__ psh1,
    const float* __restrict__ fterm2,
    float* __restrict__ kmean,
    float* __restrict__ PP2,
    const float* __restrict__ psc2, const float* __restrict__ psh2,
    float* __restrict__ k2pool,
    int mode) {
  __shared__ alignas(16) float sW1[32 * 64];
  __shared__ alignas(16) float sW2[64 * 64];
  __shared__ alignas(16) float sX[64 * XPAD];
  __shared__ float sK[4 * 16 * 64];
  __shared__ float sRS[4 * 32 * 4], sRQ[4 * 32 * 4];
  int tid = threadIdx.x, lane = tid & 31, wv = tid >> 5;

  // x_packet slab (64 rows) via async-to-LDS copies
  {
    const float* xpb = xp + (long)blockIdx.x * 64 * F_PKT;
    for (int i = tid; i < 64 * 8; i += 128) {
      int r = i >> 3, c = i & 7;
      async_copy_b128(xpb + r * F_PKT + c * 4, &sX[r * XPAD + c * 4]);
    }
  }
  // weight tiles via Tensor Data Mover (one wave issues; TENSORcnt waits)
#if __has_builtin(__builtin_amdgcn_tensor_load_to_lds)
  if (wv == 0) {
    tdm_load_2d(W1, sW1, 32, 64);
    tdm_load_2d(W2, sW2, 64, 64);
  }
#else
  for (int i = tid; i < 32 * 64; i += 128) sW1[i] = W1[i];
  for (int i = tid; i < 64 * 64; i += 128) sW2[i] = W2[i];
#endif
  async_wait0();
  if (wv == 0) tensor_wait0();

  int m0 = (blockIdx.x * 4 + wv) * 16;
  int g = m0 >> 5;
  int arow = lane & 15, khalf = (lane >> 4) << 1, col = lane & 15, rhalf = (lane >> 4) << 3;
  float* sKw = &sK[wv * 16 * 64];
  __syncthreads();

  // ---- stage 1: k_lr = lrelu(bn1(x_packet@W1 + fl1[g])) into LDS ----
  const float* ArL = &sX[(wv * 16 + arow) * XPAD];
  #pragma unroll
  for (int nt = 0; nt < 4; ++nt) {
    v8f acc = {};
    #pragma unroll
    for (int k0 = 0; k0 < 32; k0 += 4) {
      int kk = k0 + khalf;
      acc = wmma4(ArL[kk], ArL[kk + 1], sW1[kk * 64 + nt * 16 + col],
                  sW1[(kk + 1) * 64 + nt * 16 + col], acc);
    }
    int c = nt * 16 + col;
    float fb = fl1[g * 64 + c], sc = psc1[c], sh = psh1[c];
    #pragma unroll
    for (int v = 0; v < 8; ++v) {
      float x = acc[v] + fb;
      sKw[(rhalf + v) * 64 + c] = lrelu(x * sc + sh);
    }
  }
  __syncthreads();

  if (mode == 2) {  // kmean[g] += half-graph column sums / 32 (2 contributors)
    for (int cc = lane; cc < 64; cc += 32) {
      float s = 0.f;
      #pragma unroll
      for (int r = 0; r < 16; ++r) s += sKw[r * 64 + cc];
      atomicAdd(&kmean[(long)g * 64 + cc], s * (1.0f / 32.0f));
    }
  }

  // ---- stage 2: k2_raw = k_lr @ W2 + fterm2[g] ----
  v8f k2[4];
  #pragma unroll
  for (int nt = 0; nt < 4; ++nt) {
    v8f acc = {};
    #pragma unroll
    for (int k0 = 0; k0 < 64; k0 += 4) {
      int kk = k0 + khalf;
      acc = wmma4(sKw[arow * 64 + kk], sKw[arow * 64 + kk + 1],
                  sW2[kk * 64 + nt * 16 + col], sW2[(kk + 1) * 64 + nt * 16 + col], acc);
    }
    float fb = fterm2[(long)g * 64 + nt * 16 + col];
    #pragma unroll
    for (int v = 0; v < 8; ++v) acc[v] += fb;
    k2[nt] = acc;
  }

  if (mode == 2) {  // deterministic channel partials of k2_raw
    #pragma unroll
    for (int nt = 0; nt < 4; ++nt) {
      float s = 0.f, q = 0.f;
      #pragma unroll
      for (int v = 0; v < 8; ++v) { float x = k2[nt][v]; s += x; q += x * x; }
      sRS[(wv * 32 + lane) * 4 + nt] = s; sRQ[(wv * 32 + lane) * 4 + nt] = q;
    }
    __syncthreads();
    if (tid < 64) {
      int nt = tid >> 4, cl = tid & 15;
      float ts = 0.f, tq = 0.f;
      for (int w = 0; w < 4; ++w) {
        ts += sRS[(w * 32 + cl) * 4 + nt] + sRS[(w * 32 + cl + 16) * 4 + nt];
        tq += sRQ[(w * 32 + cl) * 4 + nt] + sRQ[(w * 32 + cl + 16) * 4 + nt];
      }
      PP2[blockIdx.x * 64 + tid] = ts;
      PP2[8192 * 64 + blockIdx.x * 64 + tid] = tq;
    }
  } else {          // mode 3: bn2 + lrelu, pool over 32 per graph
    __syncthreads();
    #pragma unroll
    for (int nt = 0; nt < 4; ++nt) {
      int c = nt * 16 + col;
      float sc = psc2[c], sh = psh2[c];
      #pragma unroll
      for (int v = 0; v < 8; ++v)
        sKw[(rhalf + v) * 64 + c] = lrelu(k2[nt][v] * sc + sh);
    }
    __syncthreads();
    for (int cc = lane; cc < 64; cc += 32) {
      float s = 0.f;
      #pragma unroll
      for (int r = 0; r < 16; ++r) s += sKw[r * 64 + cc];
      atomicAdd(&k2pool[(long)g * 64 + cc], s * (1.0f / 32.0f));
    }
  }
}

// ---------------- flow-side BN stats (16384 rows, 64 ch) -------------------
__global__ __launch_bounds__(256) void row_stats(const float* __restrict__ X,
                                                 float* __restrict__ PPF) {
  __shared__ float sS[256], sQ[256];
  int tid = threadIdx.x, c = tid & 63, r0 = tid >> 6;
  long base = (long)blockIdx.x * 256;
  float s = 0.f, q = 0.f;
  for (int r = r0; r < 256; r += 4) {
    float x = X[(base + r) * 64 + c];
    s += x; q += x * x;
  }
  sS[tid] = s; sQ[tid] = q;
  __syncthreads();
  if (tid < 64) {
    float ts = sS[tid] + sS[tid + 64] + sS[tid + 128] + sS[tid + 192];
    float tq = sQ[tid] + sQ[tid + 64] + sQ[tid + 128] + sQ[tid + 192];
    PPF[blockIdx.x * 64 + tid] = ts;
    PPF[64 * 64 + blockIdx.x * 64 + tid] = tq;
  }
}

// fixed-order partial reduction -> bn scale/shift (deterministic)
__global__ void bn_finalize(const float* __restrict__ PP, int NB,
                            const float* __restrict__ g, const float* __restrict__ b,
                            float n, float* __restrict__ scale, float* __restrict__ shift) {
  int c = threadIdx.x;  // 64 threads
  float s = 0.f, q = 0.f;
  for (int i = 0; i < NB; ++i) { s += PP[i * 64 + c]; q += PP[(long)NB * 64 + i * 64 + c]; }
  float m = s / n;
  float v = q / n - m * m;
  float inv = rsqrtf(v + 1e-5f);
  float sc = g[c] * inv;
  scale[c] = sc; shift[c] = b[c] - m * sc;
}

__global__ __launch_bounds__(256) void bn_lrelu(const float* __restrict__ X,
                                                const float* __restrict__ sc,
                                                const float* __restrict__ sh,
                                                float* __restrict__ Y, int n) {
  int i = blockIdx.x * 256 + threadIdx.x;
  if (i < n) {
    int c = i & 63;
    Y[i] = lrelu(X[i] * sc[c] + sh[c]);
  }
}

// -------- head tail: 64->16->8 + log_softmax (tiny, plain VALU) ------------
__global__ __launch_bounds__(256) void head_tail(const float* __restrict__ h1,
                                                 const float* __restrict__ W2,
                                                 const float* __restrict__ b2,
                                                 const float* __restrict__ W3,
                                                 const float* __restrict__ b3,
                                                 float* __restrict__ out) {
  __shared__ float sW2[64 * 16], sW3[16 * 8], sb2[16], sb3[8];
  int tid = threadIdx.x;
  for (int i = tid; i < 64 * 16; i += 256) sW2[i] = W2[i];
  if (tid < 16 * 8) sW3[tid] = W3[tid];
  if (tid < 16) sb2[tid] = b2[tid];
  if (tid < 8) sb3[tid] = b3[tid];
  __syncthreads();
  long g = (long)blockIdx.x * 256 + tid;
  const float* h = h1 + g * 64;
  float h2[16];
  #pragma unroll
  for (int j = 0; j < 16; ++j) h2[j] = sb2[j];
  for (int i = 0; i < 64; ++i) {
    float x = h[i];
    #pragma unroll
    for (int j = 0; j < 16; ++j) h2[j] += x * sW2[i * 16 + j];
  }
  float h3[8];
  #pragma unroll
  for (int j = 0; j < 8; ++j) h3[j] = sb3[j];
  for (int i = 0; i < 16; ++i) {
    float x = h2[i];
    #pragma unroll
    for (int j = 0; j < 8; ++j) h3[j] += x * sW3[i * 8 + j];
  }
  float m = h3[0];
  #pragma unroll
  for (int j = 1; j < 8; ++j) m = fmaxf(m, h3[j]);
  float se = 0.f;
  #pragma unroll
  for (int j = 0; j < 8; ++j) se += __expf(h3[j] - m);
  float lse = m + __logf(se);
  #pragma unroll
  for (int j = 0; j < 8; ++j) out[g * 8 + j] = h3[j] - lse;
}

// ---------------------------------------------------------------------------
extern "C" void kernel_launch(void* const* d_in, const int* in_sizes, int n_in,
                              void* d_out, int out_size, void* d_ws, size_t ws_size,
                              hipStream_t stream) {
  const float* x_flow   = (const float*)d_in[0];
  const float* x_pkt    = (const float*)d_in[1];
  const float* c1_pf_Wl = (const float*)d_in[2];
  const float* c1_pf_bl = (const float*)d_in[3];
  const float* c1_pf_Wr = (const float*)d_in[4];
  const float* c1_fp_Wl = (const float*)d_in[5];
  const float* c1_fp_bl = (const float*)d_in[6];
  const float* c1_fp_Wr = (const float*)d_in[7];
  const float* c2_pf_Wl = (const float*)d_in[8];
  const float* c2_pf_bl = (const float*)d_in[9];
  const float* c2_pf_Wr = (const float*)d_in[10];
  const float* c2_fp_Wl = (const float*)d_in[11];
  const float* c2_fp_bl = (const float*)d_in[12];
  const float* c2_fp_Wr = (const float*)d_in[13];
  const float* bn1f_g = (const float*)d_in[14];
  const float* bn1f_b = (const float*)d_in[15];
  const float* bn1p_g = (const float*)d_in[16];
  const float* bn1p_b = (const float*)d_in[17];
  const float* bn2f_g = (const float*)d_in[18];
  const float* bn2f_b = (const float*)d_in[19];
  const float* bn2p_g = (const float*)d_in[20];
  const float* bn2p_b = (const float*)d_in[21];
  const float* head_W1 = (const float*)d_in[22];
  const float* head_b1 = (const float*)d_in[23];
  const float* head_W2 = (const float*)d_in[24];
  const float* head_b2 = (const float*)d_in[25];
  const float* head_W3 = (const float*)d_in[26];
  const float* head_b3 = (const float*)d_in[27];
  float* out = (float*)d_out;

  // workspace layout (floats); total ~13.1M floats (~53 MB)
  float* ws = (float*)d_ws;
  float* pm     = ws;                 // 16384*32
  float* f1     = pm + 524288;        // 16384*64
  float* fl1    = f1 + 1048576;
  float* flr    = fl1 + 1048576;
  float* fterm2 = flr + 1048576;
  float* f2     = fterm2 + 1048576;
  float* f2lr   = f2 + 1048576;
  float* h1     = f2lr + 1048576;
  float* kmean  = h1 + 1048576;       // zeroed per call
  float* k2pool = kmean + 1048576;    // zeroed per call
  float* PP1    = k2pool + 1048576;   // 2*16384*64, zeroed per call
  float* PP2    = PP1 + 2097152;      // 2*8192*64
  float* PPF    = PP2 + 1048576;      // 2*64*64
  float* psc1   = PPF + 8192;
  float* psh1 = psc1 + 64; float* psc2 = psh1 + 64; float* psh2 = psc2 + 64;
  float* fsc1 = psh2 + 64; float* fsh1 = fsc1 + 64;
  float* fsc2 = fsh1 + 64; float* fsh2 = fsc2 + 64;

  // zero kmean + k2pool + PP1 (4194304 floats)
  zero_f32<<<16384, 256, 0, stream>>>(kmean, 4194304);

  // flow layer 1
  pkt_mean32<<<2048, 256, 0, stream>>>(x_pkt, pm);
  gemm_n64<<<dim3(128, 4), 256, 0, stream>>>(pm, 32, c1_pf_Wl, c1_pf_bl, f1, 32, 0);
  gemm_n64<<<dim3(128, 4), 256, 0, stream>>>(x_flow, F_FLOW, c1_pf_Wr, nullptr, f1, F_FLOW, 1);
  gemm_n64<<<dim3(128, 4), 256, 0, stream>>>(x_flow, F_FLOW, c1_fp_Wl, c1_fp_bl, fl1, F_FLOW, 0);
  row_stats<<<64, 256, 0, stream>>>(f1, PPF);
  bn_finalize<<<1, 64, 0, stream>>>(PPF, 64, bn1f_g, bn1f_b, (float)N_GRAPHS, fsc1, fsh1);
  bn_lrelu<<<4096, 256, 0, stream>>>(f1, fsc1, fsh1, flr, N_GRAPHS * HD);

  // packet layer 1 stats
  pkt_pass1<<<dim3(4096, 4), 256, 0, stream>>>(x_pkt, c1_fp_Wr, fl1, PP1);
  bn_finalize<<<1, 64, 0, stream>>>(PP1, 16384, bn1p_g, bn1p_b, (float)N_PKT, psc1, psh1);

  // layer 2 packet chain
  gemm_n64<<<dim3(128, 4), 256, 0, stream>>>(flr, 64, c2_fp_Wl, c2_fp_bl, fterm2, 64, 0);
  pkt_pass23<<<8192, 128, 0, stream>>>(x_pkt, c1_fp_Wr, c2_fp_Wr, fl1, psc1, psh1,
                                       fterm2, kmean, PP2, psc2, psh2, k2pool, 2);
  bn_finalize<<<1, 64, 0, stream>>>(PP2, 8192, bn2p_g, bn2p_b, (float)N_PKT, psc2, psh2);

  // flow layer 2
  gemm_n64<<<dim3(128, 4), 256, 0, stream>>>(kmean, 64, c2_pf_Wl, c2_pf_bl, f2, 64, 0);
  gemm_n64<<<dim3(128, 4), 256, 0, stream>>>(flr, 64, c2_pf_Wr, nullptr, f2, 64, 1);
  row_stats<<<64, 256, 0, stream>>>(f2, PPF);
  bn_finalize<<<1, 64, 0, stream>>>(PPF, 64, bn2f_g, bn2f_b, (float)N_GRAPHS, fsc2, fsh2);
  bn_lrelu<<<4096, 256, 0, stream>>>(f2, fsc2, fsh2, f2lr, N_GRAPHS * HD);

  // packet layer 2 apply + pool
  pkt_pass23<<<8192, 128, 0, stream>>>(x_pkt, c1_fp_Wr, c2_fp_Wr, fl1, psc1, psh1,
                                       fterm2, kmean, PP2, psc2, psh2, k2pool, 3);

  // head: h1 = [f2lr | k2pool] @ W1 + b1  (split K over the two halves)
  gemm_n64<<<dim3(128, 4), 256, 0, stream>>>(f2lr, 64, head_W1, head_b1, h1, 64, 0);
  gemm_n64<<<dim3(128, 4), 256, 0, stream>>>(k2pool, 64, head_W1 + 64 * 64, nullptr, h1, 64, 1);
  head_tail<<<64, 256, 0, stream>>>(h1, head_W2, head_b2, head_W3, head_b3, out);

  (void)in_sizes; (void)n_in; (void)out_size; (void)ws_size;
}